// MSSLoss_19104014533453
// MI455X (gfx1250) — compile-verified
//
#include <hip/hip_runtime.h>

// CDNA5 / gfx1250: wave32, WMMA 16x16 shapes. We use V_WMMA_F32_16X16X4_F32
// for the separable 11-tap Gaussian passes expressed as banded matmuls,
// keeping all statistics in fp32 (sigma = E[x^2]-mu^2 is cancellation-prone).

typedef __attribute__((ext_vector_type(2))) float v2f;
typedef __attribute__((ext_vector_type(8))) float v8f;

#define C1F 0.0001f   // 0.01^2
#define C2F 0.0009f   // 0.03^2

__device__ __forceinline__ v8f wmma_f32_16x16x4(v2f a, v2f b, v8f c) {
  // 8-arg form: (neg_a, A, neg_b, B, c_mod, C, reuse_a, reuse_b)
  return __builtin_amdgcn_wmma_f32_16x16x4_f32(false, a, false, b, (short)0, c,
                                               false, false);
}

// One wave per 16x16 output tile per channel.
// Computes per-tile partial sums of sim and cs over the valid (H-10)^2 region.
__global__ __launch_bounds__(32) void msssim_stats_kernel(
    const float* __restrict__ pin, const float* __restrict__ tin,
    float* __restrict__ part, int H, int ntiles) {
  const int W = H, OH = H - 10;
  const int TX = (OH + 15) >> 4;
  const int tileX = blockIdx.x, tileY = blockIdx.y, ch = blockIdx.z;
  const int lane = threadIdx.x;
  const int n = lane & 15;    // col (B/C/D) or row (A) index per WMMA layout
  const int hi = lane >> 4;   // lane group: K-pair selector for A/B fragments
  const int r0 = tileY << 4, c0 = tileX << 4;

  __shared__ float sG[16];         // normalized gaussian, padded with zeros
  __shared__ float sP[26 * 28];    // p patch, 2-col zero pad for K=28
  __shared__ float sT[26 * 28];    // t patch
  __shared__ float sH[5][28 * 16]; // horizontal-pass results, K-padded rows

  // ---- Gaussian window (WIN=11, SIGMA=1.5), normalized ----
  {
    float gsum = 0.f;
#pragma unroll
    for (int j = 0; j < 11; ++j) {
      float x = (float)(j - 5);
      gsum += __expf(-(x * x) * (1.f / 4.5f));
    }
    if (lane < 16) {
      float gv = 0.f;
      if (lane < 11) {
        float x = (float)(lane - 5);
        gv = __expf(-(x * x) * (1.f / 4.5f)) / gsum;
      }
      sG[lane] = gv;
    }
  }

  // ---- Stage 26x26 input patches (clamped reads; clamped region only feeds
  //      masked-out outputs, so results for valid pixels are exact) ----
  const float* pc = pin + (size_t)ch * H * W;
  const float* tc = tin + (size_t)ch * H * W;
  for (int i = lane; i < 26 * 26; i += 32) {
    int rr = i / 26, cc = i - rr * 26;
    int gr = min(r0 + rr, H - 1);
    int gc = min(c0 + cc, W - 1);
    size_t off = (size_t)gr * W + gc;
    sP[rr * 28 + cc] = pc[off];
    sT[rr * 28 + cc] = tc[off];
  }
  if (lane < 26) {
    sP[lane * 28 + 26] = 0.f; sP[lane * 28 + 27] = 0.f;
    sT[lane * 28 + 26] = 0.f; sT[lane * 28 + 27] = 0.f;
  }
  {
    int i = lane; // zero K-pad rows 26,27 of sH (32 lanes cover 2*16)
#pragma unroll
    for (int s = 0; s < 5; ++s) sH[s][26 * 16 + i] = 0.f;
  }
  __syncthreads();

  // ---- Banded Gaussian fragments.
  // Horizontal B[k][j] = g[k-j] (j = lane&15) and vertical A[m][k] = g[k-m]
  // (m = lane&15) have identical per-lane values under the 16x4 A / 4x16 B
  // layouts (per lane, VGPR pair holds K = 2*hi, 2*hi+1), so one array serves
  // both passes. Out-of-band -> 0.
  v2f wb[7];
#pragma unroll
  for (int kk = 0; kk < 7; ++kk) {
    int k0 = 4 * kk + 2 * hi;
    int i0 = k0 - n, i1 = k0 + 1 - n;
    float g0 = sG[i0 & 15]; g0 = ((unsigned)i0 < 11u) ? g0 : 0.f;
    float g1 = sG[i1 & 15]; g1 = ((unsigned)i1 < 11u) ? g1 : 0.f;
    wb[kk].x = g0; wb[kk].y = g1;
  }

  const v8f z8 = {0.f, 0.f, 0.f, 0.f, 0.f, 0.f, 0.f, 0.f};

  // ---- Horizontal pass: H[r][j] = sum_q g[q]*In[r][j+q], rows 0..25.
  // Two 16-row blocks (rb=0 and rb=10, overlap rows recomputed). 5 signals
  // share the A-data loads; 5 independent WMMA accumulation chains give ILP.
#pragma unroll
  for (int rbI = 0; rbI < 2; ++rbI) {
    const int rb = rbI * 10;
    v8f hc[5] = {z8, z8, z8, z8, z8};
#pragma unroll
    for (int kk = 0; kk < 7; ++kk) {
      int k0 = 4 * kk + 2 * hi;
      const float* prow = &sP[(rb + n) * 28 + k0];
      const float* trow = &sT[(rb + n) * 28 + k0];
      float px = prow[0], py = prow[1];
      float tx = trow[0], ty = trow[1];
      v2f b = wb[kk];
      v2f a;
      a.x = px;      a.y = py;      hc[0] = wmma_f32_16x16x4(a, b, hc[0]);
      a.x = tx;      a.y = ty;      hc[1] = wmma_f32_16x16x4(a, b, hc[1]);
      a.x = px * px; a.y = py * py; hc[2] = wmma_f32_16x16x4(a, b, hc[2]);
      a.x = tx * tx; a.y = ty * ty; hc[3] = wmma_f32_16x16x4(a, b, hc[3]);
      a.x = px * tx; a.y = py * ty; hc[4] = wmma_f32_16x16x4(a, b, hc[4]);
    }
    // D layout: VGPR v holds row v (lanes 0-15) / v+8 (lanes 16-31), col n.
#pragma unroll
    for (int v = 0; v < 8; ++v) {
      int row = rb + v + 8 * hi;
      bool doStore = (rbI == 0) || (row >= 16);
      if (doStore) {
#pragma unroll
        for (int s = 0; s < 5; ++s) sH[s][row * 16 + n] = hc[s][v];
      }
    }
  }
  __syncthreads();

  // ---- Vertical pass: Out[m][j] = sum_k g[k-m] * H[k][j], K padded to 28.
  v8f o[5] = {z8, z8, z8, z8, z8};
#pragma unroll
  for (int kk = 0; kk < 7; ++kk) {
    int k0 = 4 * kk + 2 * hi;
    v2f a = wb[kk];
#pragma unroll
    for (int s = 0; s < 5; ++s) {
      v2f b;
      b.x = sH[s][k0 * 16 + n];
      b.y = sH[s][(k0 + 1) * 16 + n];
      o[s] = wmma_f32_16x16x4(a, b, o[s]);
    }
  }

  // ---- Per-pixel SSIM/CS and masked tile accumulation ----
  float simSum = 0.f, csSum = 0.f;
#pragma unroll
  for (int v = 0; v < 8; ++v) {
    int orow = r0 + v + 8 * hi;
    int ocol = c0 + n;
    float mu1 = o[0][v], mu2 = o[1][v];
    float e11 = o[2][v], e22 = o[3][v], e12 = o[4][v];
    float mu1s = mu1 * mu1, mu2s = mu2 * mu2, mu12 = mu1 * mu2;
    float v1 = 2.f * (e12 - mu12) + C2F;
    float v2_ = (e11 - mu1s) + (e22 - mu2s) + C2F;
    float cs = v1 / v2_;
    float sim = ((2.f * mu12 + C1F) * v1) / ((mu1s + mu2s + C1F) * v2_);
    if (orow < OH && ocol < OH) { simSum += sim; csSum += cs; }
  }
#pragma unroll
  for (int off = 16; off > 0; off >>= 1) {
    simSum += __shfl_down(simSum, off);
    csSum  += __shfl_down(csSum, off);
  }
  if (lane == 0) {
    int tile = tileY * TX + tileX;
    part[(size_t)(0 * 10 + ch) * ntiles + tile] = simSum;
    part[(size_t)(1 * 10 + ch) * ntiles + tile] = csSum;
  }
}

// Deterministic fixed-order reduction of per-tile partials into acc[2][5][10].
__global__ __launch_bounds__(256) void reduce_parts_kernel(
    const float* __restrict__ part, float* __restrict__ acc, int ntiles,
    int level) {
  int ch = blockIdx.x, val = blockIdx.y;
  const float* src = part + (size_t)(val * 10 + ch) * ntiles;
  float s = 0.f;
  for (int i = threadIdx.x; i < ntiles; i += 256) s += src[i];
  __shared__ float red[256];
  red[threadIdx.x] = s;
  __syncthreads();
  for (int o = 128; o > 0; o >>= 1) {
    if ((int)threadIdx.x < o) red[threadIdx.x] += red[threadIdx.x + o];
    __syncthreads();
  }
  if (threadIdx.x == 0) acc[val * 50 + level * 10 + ch] = red[0];
}

// 2x2 average pool, NCHW with C=10, square images.
__global__ __launch_bounds__(256) void avgpool2_kernel(
    const float* __restrict__ in, float* __restrict__ out, int Ho) {
  int Wi = Ho * 2;
  long long total = 10LL * Ho * Ho;
  long long i = (long long)blockIdx.x * 256 + threadIdx.x;
  if (i >= total) return;
  int hw = Ho * Ho;
  int ch = (int)(i / hw);
  int rem = (int)(i - (long long)ch * hw);
  int y = rem / Ho, x = rem - y * Ho;
  const float* b = in + (size_t)ch * Wi * Wi + (size_t)(2 * y) * Wi + 2 * x;
  out[i] = 0.25f * (b[0] + b[1] + b[Wi] + b[Wi + 1]);
}

// acc layout: [0..49] sim sums (level*10+ch), [50..99] cs sums.
__global__ void finalize_kernel(const float* __restrict__ acc,
                                float* __restrict__ out) {
  if (threadIdx.x != 0 || blockIdx.x != 0) return;
  const float w[5] = {0.0448f, 0.2856f, 0.3001f, 0.2363f, 0.1333f};
  float total = 0.f;
  int H4 = 1024 >> 4;
  float cnt4 = (float)((H4 - 10) * (H4 - 10));
  for (int c = 0; c < 10; ++c) {
    float mssim4 = acc[4 * 10 + c] / cnt4;
    float p2last = powf(mssim4, w[4]);
    float prod = 1.f;
    for (int l = 0; l < 4; ++l) {
      int Hl = 1024 >> l;
      float cnt = (float)((Hl - 10) * (Hl - 10));
      float mcs = acc[50 + l * 10 + c] / cnt;
      // reference: prod over l<4 of (mcs[l]^w[l] * mssim[4]^w[4])
      prod *= powf(mcs, w[l]) * p2last;
    }
    total += prod;
  }
  out[0] = 1.f - total;
}

extern "C" void kernel_launch(void* const* d_in, const int* in_sizes, int n_in,
                              void* d_out, int out_size, void* d_ws,
                              size_t ws_size, hipStream_t stream) {
  (void)in_sizes; (void)n_in; (void)out_size; (void)ws_size;
  const float* p0 = (const float*)d_in[0];
  const float* t0 = (const float*)d_in[1];
  float* ws = (float*)d_ws;

  // Workspace layout (floats):
  //   [0,128)            : acc[2][5][10] (padded)
  //   [128, 128+81920)   : per-tile partials (max 2*10*4096)
  //   [82048, ...)       : pyramid buffers p1,t1,p2,t2,p3,t3,p4,t4 (~27.9 MB)
  float* acc = ws;
  float* part = ws + 128;
  float* buf = ws + 128 + 81920;
  size_t off = 0;

  const float* Pl = p0;
  const float* Tl = t0;
  for (int l = 0; l < 5; ++l) {
    int H = 1024 >> l;
    int OH = H - 10;
    int TX = (OH + 15) >> 4;
    int ntiles = TX * TX;
    dim3 grid(TX, TX, 10);
    msssim_stats_kernel<<<grid, 32, 0, stream>>>(Pl, Tl, part, H, ntiles);
    reduce_parts_kernel<<<dim3(10, 2), 256, 0, stream>>>(part, acc, ntiles, l);
    if (l < 4) {
      int Ho = H / 2;
      size_t elems = (size_t)10 * Ho * Ho;
      float* Pn = buf + off; off += elems;
      float* Tn = buf + off; off += elems;
      int blocks = (int)((elems + 255) / 256);
      avgpool2_kernel<<<blocks, 256, 0, stream>>>(Pl, Pn, Ho);
      avgpool2_kernel<<<blocks, 256, 0, stream>>>(Tl, Tn, Ho);
      Pl = Pn; Tl = Tn;
    }
  }
  finalize_kernel<<<1, 32, 0, stream>>>(acc, (float*)d_out);
}